// TopkQuantLayer_57947698757888
// MI455X (gfx1250) — compile-verified
//
#include <hip/hip_runtime.h>

// MI455X (gfx1250) top-k + partitioned min/max quant-dequant.
// Memory-bound (~1.07 GB traffic -> ~46us @ 23.3 TB/s). No GEMM structure, so
// no WMMA; CDNA5 paths used: gfx1250 async global->LDS streaming (ASYNCcnt
// depth-4 ring, s_wait_asynccnt) in every full-array pass, wave32 LDS
// histograms (ds atomics), b128 loads, global_prefetch_b8 fallback.

typedef unsigned int u32;

#define BLK 256
#define GRD 2048

// ---------------- workspace layout (u32 words) ----------------
#define WS_MAGH1   0u        // 2048 : |x| key bits[30:20]
#define WS_MAGH2   2048u     // 1024 : bits[19:10] within selected bin
#define WS_MAGH3   3072u     // 1024 : bits[9:0]   within selected bin
#define WS_CTRL    4096u     // 16   : [0]=bin1 [1]=rem1 [2]=top21 [3]=rem2 [4]=tmag [5]=budget [6]=tieCnt
#define WS_QH1     4112u     // 4096 : signed key bits[31:20] of kept values
#define WS_QTB1    8208u     // 16   : per-rank target top-12 bin
#define WS_QRR1    8224u     // 16   : per-rank residual rank
#define WS_QH2     8240u     // 16*1024
#define WS_QTB12   24624u    // 16   : per-rank target top-22 bits
#define WS_QRR2    24640u    // 16
#define WS_QH3     24656u    // 16*1024
#define WS_UKEY    41040u    // 16   : final 32-bit keys of the 16 order stats
#define WS_BKEY    41056u    // 7    : partition boundary keys
#define WS_FPARAM  41064u    // 24 floats : mn[8], mx[8], step[8]
#define WS_TOTAL   41088u    // ~164 KB

// gfx1250 async global->LDS path (probe via __has_builtin; safe fallback).
#if defined(__has_builtin)
#if __has_builtin(__builtin_amdgcn_global_load_async_to_lds_b128) && \
    __has_builtin(__builtin_amdgcn_s_wait_asynccnt)
#define HAVE_ASYNC_LDS 1
#endif
#endif

#define COMPILER_FENCE() asm volatile("" ::: "memory")

// monotone ascending key for signed float bit pattern
static __device__ __forceinline__ u32 sortkey(u32 bits) {
  return bits ^ ((u32)((int)bits >> 31) | 0x80000000u);
}
// inverse transform
static __device__ __forceinline__ u32 unkey(u32 u) {
  return (u & 0x80000000u) ? (u ^ 0x80000000u) : ~u;
}

#if HAVE_ASYNC_LDS
// Builtin prototype (from the round-1 diagnostic): arg0 is a NON-const pointer
// to a native int 4-vector in AS(1); arg1 is the LDS destination in AS(3).
typedef __attribute__((__vector_size__(4 * sizeof(int)))) int v4i;
static __device__ __forceinline__ void async_cp16(const uint4* g, uint4* l) {
  __builtin_amdgcn_global_load_async_to_lds_b128(
      (__attribute__((address_space(1))) v4i*)g,
      (__attribute__((address_space(3))) v4i*)l, 0, 0);
}
#endif

// Grid-stride streaming over uint4 elements through a depth-4 async->LDS ring.
// ASYNCcnt tracks in-flight tiles (async loads complete in order, ISA §4.1), so
// a wave keeps 4 tiles (64B/lane) in flight with zero VGPR pressure. Whenever
// the steady-state branch is taken, all 4 covering issues have occurred, so
// s_wait_asynccnt(3) completes the oldest (current) tile; tail drains with 0.
template <typename Body>
static __device__ __forceinline__ void stream4(const uint4* __restrict__ x, u32 nvec, Body body) {
  const u32 stride = gridDim.x * blockDim.x;
  const u32 i0 = blockIdx.x * blockDim.x + threadIdx.x;
#if HAVE_ASYNC_LDS
  __shared__ uint4 stage[4][BLK];
#pragma unroll
  for (u32 t = 0; t < 4u; ++t) {
    const u32 idx = i0 + t * stride;
    if (idx < nvec) async_cp16(&x[idx], &stage[t][threadIdx.x]);
  }
  u32 slot = 0;
  for (u32 i = i0; i < nvec; i += stride) {
    if (i + 3u * stride < nvec) __builtin_amdgcn_s_wait_asynccnt(3);
    else                        __builtin_amdgcn_s_wait_asynccnt(0);
    COMPILER_FENCE();
    const uint4 v = stage[slot][threadIdx.x];   // each lane reads its own slot
    body(i, v);                                  // consumes v before slot reuse
    COMPILER_FENCE();
    const u32 rep = i + 4u * stride;
    if (rep < nvec) async_cp16(&x[rep], &stage[slot][threadIdx.x]);
    slot = (slot + 1u) & 3u;
  }
#else
  for (u32 i = i0; i < nvec; i += stride) {
    const uint4 v = x[i];
    __builtin_prefetch(&x[i + stride], 0, 0);
    body(i, v);
  }
#endif
}

// One wave (32 lanes) locates the bin containing ascending 0-based `rank` in an
// nbins histogram. Returns true on the (single) hit lane with outputs set.
static __device__ __forceinline__ bool wave_find_rank(const u32* __restrict__ h,
                                                      u32 nbins, u32 rank,
                                                      u32* bin, u32* rem, u32* eq) {
  const u32 lane = threadIdx.x & 31u;
  const u32 bpl = nbins >> 5;           // bins per lane (contiguous chunk)
  const u32 base = lane * bpl;
  u32 s = 0;
  for (u32 b = 0; b < bpl; ++b) s += h[base + b];
  u32 inc = s;
  for (u32 d = 1; d < 32; d <<= 1) {
    u32 y = __shfl_up(inc, d, 32);
    if (lane >= d) inc += y;
  }
  u32 exc = inc - s;
  bool hit = (rank >= exc) && (rank < exc + s);
  if (hit) {
    u32 local = rank - exc, cum = 0;
    for (u32 b = 0; b < bpl; ++b) {
      u32 cb = h[base + b];
      if (local < cum + cb) { *bin = base + b; *rem = local - cum; *eq = cb; return true; }
      cum += cb;
    }
  }
  return false;
}

static __device__ __forceinline__ void hist_mag4(u32* lh, uint4 v) {
  atomicAdd(&lh[(v.x & 0x7FFFFFFFu) >> 20], 1u);
  atomicAdd(&lh[(v.y & 0x7FFFFFFFu) >> 20], 1u);
  atomicAdd(&lh[(v.z & 0x7FFFFFFFu) >> 20], 1u);
  atomicAdd(&lh[(v.w & 0x7FFFFFFFu) >> 20], 1u);
}

// ---------------- pass M1: 2048-bin LDS histogram of |x| bits[30:20] ----------------
__global__ void k_hist_mag1(const uint4* __restrict__ x, u32 nvec, u32* __restrict__ ws) {
  __shared__ u32 lh[2048];
  for (u32 j = threadIdx.x; j < 2048; j += blockDim.x) lh[j] = 0;
  __syncthreads();
  stream4(x, nvec, [&](u32, uint4 v) { hist_mag4(lh, v); });
  __syncthreads();
  u32* gh = ws + WS_MAGH1;
  for (u32 j = threadIdx.x; j < 2048; j += blockDim.x) {
    u32 cv = lh[j];
    if (cv) atomicAdd(&gh[j], cv);
  }
}

__global__ void k_res_m1(u32* ws, u32 rank) {   // rank = n - k (ascending)
  u32 b, r, e;
  if (wave_find_rank(ws + WS_MAGH1, 2048, rank, &b, &r, &e)) {
    ws[WS_CTRL + 0] = b;
    ws[WS_CTRL + 1] = r;
  }
}

__global__ void k_hist_mag2(const uint4* __restrict__ x, u32 nvec, u32* __restrict__ ws) {
  const u32 bin1 = ws[WS_CTRL + 0];
  u32* __restrict__ h2 = ws + WS_MAGH2;
  stream4(x, nvec, [&](u32, uint4 v) {
    u32 a[4] = {v.x, v.y, v.z, v.w};
#pragma unroll
    for (int c = 0; c < 4; ++c) {
      u32 m = a[c] & 0x7FFFFFFFu;
      if ((m >> 20) == bin1) atomicAdd(&h2[(m >> 10) & 1023u], 1u);
    }
  });
}

__global__ void k_res_m2(u32* ws) {
  const u32 bin1 = ws[WS_CTRL + 0];
  const u32 rank = ws[WS_CTRL + 1];
  u32 b, r, e;
  if (wave_find_rank(ws + WS_MAGH2, 1024, rank, &b, &r, &e)) {
    ws[WS_CTRL + 2] = (bin1 << 10) | b;
    ws[WS_CTRL + 3] = r;
  }
}

__global__ void k_hist_mag3(const uint4* __restrict__ x, u32 nvec, u32* __restrict__ ws) {
  const u32 top21 = ws[WS_CTRL + 2];
  u32* __restrict__ h3 = ws + WS_MAGH3;
  stream4(x, nvec, [&](u32, uint4 v) {
    u32 a[4] = {v.x, v.y, v.z, v.w};
#pragma unroll
    for (int c = 0; c < 4; ++c) {
      u32 m = a[c] & 0x7FFFFFFFu;
      if ((m >> 10) == top21) atomicAdd(&h3[m & 1023u], 1u);
    }
  });
}

__global__ void k_res_m3(u32* ws) {
  const u32 top21 = ws[WS_CTRL + 2];
  const u32 rank = ws[WS_CTRL + 3];
  u32 b, r, e;
  if (wave_find_rank(ws + WS_MAGH3, 1024, rank, &b, &r, &e)) {
    ws[WS_CTRL + 4] = (top21 << 10) | b;  // tmag: exact 31-bit |x| key of k-th largest
    ws[WS_CTRL + 5] = e - r;              // budget: how many tied-at-threshold to keep
  }
}

// ---------------- pass Q1: 4096-bin LDS histogram of kept signed keys bits[31:20] ----------------
__global__ void k_hist_q1(const uint4* __restrict__ x, u32 nvec, u32* __restrict__ ws) {
  __shared__ u32 lh[4096];
  const u32 tmag = ws[WS_CTRL + 4];
  for (u32 j = threadIdx.x; j < 4096; j += blockDim.x) lh[j] = 0;
  __syncthreads();
  stream4(x, nvec, [&](u32, uint4 v) {
    u32 a[4] = {v.x, v.y, v.z, v.w};
#pragma unroll
    for (int c = 0; c < 4; ++c)
      if ((a[c] & 0x7FFFFFFFu) >= tmag) atomicAdd(&lh[sortkey(a[c]) >> 20], 1u);
  });
  __syncthreads();
  u32* gh = ws + WS_QH1;
  for (u32 j = threadIdx.x; j < 4096; j += blockDim.x) {
    u32 cv = lh[j];
    if (cv) atomicAdd(&gh[j], cv);
  }
}

// 16 waves, wave w locates ascending rank R(w) in the shared 4096-bin histogram.
// R = {0, c-1, c, 2c-1, ..., 7c, 8c-1}: exactly the mn/mx order statistics.
__global__ void k_res_q1(u32* ws, u32 k) {
  const u32 w = threadIdx.x >> 5;
  const u32 c = k >> 3;
  const u32 rank = (w & 1u) ? (((w + 1u) >> 1) * c - 1u) : ((w >> 1) * c);
  u32 b, r, e;
  if (wave_find_rank(ws + WS_QH1, 4096, rank, &b, &r, &e)) {
    ws[WS_QTB1 + w] = b;
    ws[WS_QRR1 + w] = r;
  }
}

__global__ void k_hist_q2(const uint4* __restrict__ x, u32 nvec, u32* __restrict__ ws) {
  const u32 tmag = ws[WS_CTRL + 4];
  u32 tb[16];
#pragma unroll
  for (int j = 0; j < 16; ++j) tb[j] = ws[WS_QTB1 + j];
  u32* __restrict__ h2 = ws + WS_QH2;
  stream4(x, nvec, [&](u32, uint4 v) {
    u32 a[4] = {v.x, v.y, v.z, v.w};
#pragma unroll
    for (int c = 0; c < 4; ++c) {
      if ((a[c] & 0x7FFFFFFFu) >= tmag) {
        u32 u = sortkey(a[c]);
        u32 t12 = u >> 20, mid = (u >> 10) & 1023u;
#pragma unroll
        for (int j = 0; j < 16; ++j)
          if (t12 == tb[j]) atomicAdd(&h2[(u32)j * 1024u + mid], 1u);
      }
    }
  });
}

__global__ void k_res_q2(u32* ws) {
  const u32 w = threadIdx.x >> 5;
  const u32 tb1 = ws[WS_QTB1 + w];
  const u32 rank = ws[WS_QRR1 + w];
  u32 b, r, e;
  if (wave_find_rank(ws + WS_QH2 + w * 1024u, 1024, rank, &b, &r, &e)) {
    ws[WS_QTB12 + w] = (tb1 << 10) | b;
    ws[WS_QRR2 + w] = r;
  }
}

__global__ void k_hist_q3(const uint4* __restrict__ x, u32 nvec, u32* __restrict__ ws) {
  const u32 tmag = ws[WS_CTRL + 4];
  u32 tb[16];
#pragma unroll
  for (int j = 0; j < 16; ++j) tb[j] = ws[WS_QTB12 + j];
  u32* __restrict__ h3 = ws + WS_QH3;
  stream4(x, nvec, [&](u32, uint4 v) {
    u32 a[4] = {v.x, v.y, v.z, v.w};
#pragma unroll
    for (int c = 0; c < 4; ++c) {
      if ((a[c] & 0x7FFFFFFFu) >= tmag) {
        u32 u = sortkey(a[c]);
        u32 t22 = u >> 10, lo = u & 1023u;
#pragma unroll
        for (int j = 0; j < 16; ++j)
          if (t22 == tb[j]) atomicAdd(&h3[(u32)j * 1024u + lo], 1u);
      }
    }
  });
}

// Resolve last 10 bits of the 16 order stats, then build quant params.
__global__ void k_finalize(u32* ws) {
  const u32 w = threadIdx.x >> 5;
  const u32 t22 = ws[WS_QTB12 + w];
  const u32 rank = ws[WS_QRR2 + w];
  u32 b, r, e;
  if (wave_find_rank(ws + WS_QH3 + w * 1024u, 1024, rank, &b, &r, &e))
    ws[WS_UKEY + w] = (t22 << 10) | b;
  __syncthreads();
  if (threadIdx.x == 0) {
    u32 uk[16];
    float V[16];
#pragma unroll
    for (int j = 0; j < 16; ++j) {
      uk[j] = ws[WS_UKEY + j];
      V[j] = __uint_as_float(unkey(uk[j]));   // ascending values
    }
    float* fp = (float*)(ws + WS_FPARAM);
#pragma unroll
    for (int p = 0; p < 8; ++p) {
      float mn = V[2 * (7 - p)];     // ascending rank (7-p)*c
      float mx = V[15 - 2 * p];      // ascending rank (8-p)*c - 1
      fp[p] = mn;
      fp[8 + p] = mx;
      fp[16 + p] = (mx - mn) / 255.0f;
    }
#pragma unroll
    for (int j = 1; j <= 7; ++j)     // boundary keys = mx of partitions 1..7
      ws[WS_BKEY + (j - 1)] = uk[15 - 2 * j];
  }
}

static __device__ __forceinline__ float quant_one(u32 bits, u32 tmag, u32 budget, u32* tie,
                                                  const u32* sbk, const float* smn,
                                                  const float* smx, const float* sst) {
  u32 m = bits & 0x7FFFFFFFu;
  if (m < tmag) return 0.0f;
  if (m == tmag) {
    if (atomicAdd(tie, 1u) >= budget) return 0.0f;
  }
  u32 u = sortkey(bits);
  u32 p = 0;
#pragma unroll
  for (int j = 0; j < 7; ++j) p += (sbk[j] >= u) ? 1u : 0u;
  float v = __uint_as_float(bits);
  float mn = smn[p], mx = smx[p];
  if (mn == mx) return v;            // reference passthrough for degenerate chunk
  float s = sst[p];
  return rintf((v - mn) / s) * s + mn;   // round-half-even, matches jnp.round
}

__global__ void k_scatter(const uint4* __restrict__ x, float4* __restrict__ out,
                          u32 nvec, u32* __restrict__ ws) {
  __shared__ float smn[8], smx[8], sst[8];
  __shared__ u32 sbk[7];
  const float* fp = (const float*)(ws + WS_FPARAM);
  if (threadIdx.x < 8) {
    smn[threadIdx.x] = fp[threadIdx.x];
    smx[threadIdx.x] = fp[8 + threadIdx.x];
    sst[threadIdx.x] = fp[16 + threadIdx.x];
  }
  if (threadIdx.x < 7) sbk[threadIdx.x] = ws[WS_BKEY + threadIdx.x];
  __syncthreads();
  const u32 tmag = ws[WS_CTRL + 4];
  const u32 budget = ws[WS_CTRL + 5];
  u32* tie = ws + WS_CTRL + 6;
  stream4(x, nvec, [&](u32 i, uint4 v) {
    float4 o;
    o.x = quant_one(v.x, tmag, budget, tie, sbk, smn, smx, sst);
    o.y = quant_one(v.y, tmag, budget, tie, sbk, smn, smx, sst);
    o.z = quant_one(v.z, tmag, budget, tie, sbk, smn, smx, sst);
    o.w = quant_one(v.w, tmag, budget, tie, sbk, smn, smx, sst);
    out[i] = o;
  });
}

extern "C" void kernel_launch(void* const* d_in, const int* in_sizes, int n_in,
                              void* d_out, int out_size, void* d_ws, size_t ws_size,
                              hipStream_t stream) {
  (void)n_in; (void)out_size; (void)ws_size;
  const u32 n = (u32)in_sizes[0];
  const u32 k = n / 4;       // RATIO = 0.25
  const u32 nvec = n / 4;    // processed as uint4 (n = 2^25)
  const uint4* x = (const uint4*)d_in[0];
  float4* out = (float4*)d_out;
  u32* ws = (u32*)d_ws;

  (void)hipMemsetAsync(d_ws, 0, (size_t)WS_TOTAL * sizeof(u32), stream);

  k_hist_mag1<<<GRD, BLK, 0, stream>>>(x, nvec, ws);
  k_res_m1<<<1, 32, 0, stream>>>(ws, n - k);
  k_hist_mag2<<<GRD, BLK, 0, stream>>>(x, nvec, ws);
  k_res_m2<<<1, 32, 0, stream>>>(ws);
  k_hist_mag3<<<GRD, BLK, 0, stream>>>(x, nvec, ws);
  k_res_m3<<<1, 32, 0, stream>>>(ws);
  k_hist_q1<<<GRD, BLK, 0, stream>>>(x, nvec, ws);
  k_res_q1<<<1, 512, 0, stream>>>(ws, k);
  k_hist_q2<<<GRD, BLK, 0, stream>>>(x, nvec, ws);
  k_res_q2<<<1, 512, 0, stream>>>(ws);
  k_hist_q3<<<GRD, BLK, 0, stream>>>(x, nvec, ws);
  k_finalize<<<1, 512, 0, stream>>>(ws);
  k_scatter<<<GRD, BLK, 0, stream>>>(x, out, nvec, ws);
}